// StaticGNN_44229573214310
// MI455X (gfx1250) — compile-verified
//
#include <hip/hip_runtime.h>
#include <hip/hip_bf16.h>

typedef float v2f __attribute__((ext_vector_type(2)));
typedef float v8f __attribute__((ext_vector_type(8)));

#define THREADS 256
#define WAVES_PER_BLOCK 8   // 256 threads / wave32

// ---------------------------------------------------------------- degrees
__global__ void deg_init_kernel(float* __restrict__ deg, int n) {
    int i = blockIdx.x * blockDim.x + threadIdx.x;
    if (i < n) deg[i] = 1.0f;                      // self-loop contribution
}

__global__ void deg_accum_kernel(const int* __restrict__ dst,
                                 float* __restrict__ deg, int e) {
    int i = blockIdx.x * blockDim.x + threadIdx.x;
    if (i < e) atomicAdd(&deg[dst[i]], 1.0f);
}

__global__ void rsqrt_kernel(float* __restrict__ deg, int n) {
    int i = blockIdx.x * blockDim.x + threadIdx.x;
    if (i < n) deg[i] = rsqrtf(deg[i]);            // deg >= 1 always (self-loop)
}

// ---------------------------------------------------------------- WMMA GEMM
// Y[N x 20] = X[N x K] @ W[K x 20]   (fp32, V_WMMA_F32_16X16X4_F32)
// One wave computes one 16-row tile; two 16x16 accumulators cover cols 0..31,
// W padded to 32 columns in LDS with zeros so cols 20..31 contribute nothing.
template <int K>
__global__ __launch_bounds__(THREADS)
void wmma_gemm_kernel(const float* __restrict__ X, const float* __restrict__ W,
                      float* __restrict__ Y, int nRows, int nTiles) {
    __shared__ float sW[K * 32];
    const int tid = threadIdx.x;
    for (int i = tid; i < K * 32; i += THREADS) {
        int k = i >> 5, n = i & 31;
        sW[i] = (n < 20) ? W[k * 20 + n] : 0.0f;
    }
    __syncthreads();                                // only block-wide barrier

    const int wave = tid >> 5;
    const int lane = tid & 31;
    const int tile = blockIdx.x * WAVES_PER_BLOCK + wave;
    if (tile >= nTiles) return;

    const int row0 = tile * 16;
    const int m    = lane & 15;        // row within tile (A) / col within tile (B)
    const int koff = (lane >> 4) * 2;  // K-pair selector per ISA 16x4 layout

    int arow = row0 + m;
    if (arow >= nRows) arow = nRows - 1;            // clamp (stores guarded)
    const float* xrow = X + (size_t)arow * K;

    v8f acc0 = {0.f, 0.f, 0.f, 0.f, 0.f, 0.f, 0.f, 0.f};
    v8f acc1 = {0.f, 0.f, 0.f, 0.f, 0.f, 0.f, 0.f, 0.f};

#pragma unroll 4
    for (int k0 = 0; k0 < K; k0 += 4) {
        const int kb = k0 + koff;
        v2f a;                                      // A: 16x4 fp32 fragment
        a.x = xrow[kb];
        a.y = xrow[kb + 1];
        v2f b0, b1;                                 // B: 4x16 fp32 fragments
        b0.x = sW[kb * 32 + m];
        b0.y = sW[(kb + 1) * 32 + m];
        b1.x = sW[kb * 32 + 16 + m];
        b1.y = sW[(kb + 1) * 32 + 16 + m];
        acc0 = __builtin_amdgcn_wmma_f32_16x16x4_f32(false, a, false, b0,
                                                     (short)0, acc0, false, false);
        acc1 = __builtin_amdgcn_wmma_f32_16x16x4_f32(false, a, false, b1,
                                                     (short)0, acc1, false, false);
    }

    // D layout: VGPR r, lanes 0-15 -> M=r, lanes 16-31 -> M=r+8, N=lane&15
    const int half = lane >> 4;
#pragma unroll
    for (int r = 0; r < 8; ++r) {
        int row = row0 + r + half * 8;
        if (row < nRows) {
            float* yr = Y + (size_t)row * 20;
            yr[m] = acc0[r];                        // cols 0..15
            if (m < 4) yr[16 + m] = acc1[r];        // cols 16..19 (20..31 dropped)
        }
    }
}

// ---------------------------------------------------------------- aggregation
// out[i,c] = bias[c] + h[i,c]*dinv[i]^2   (bias + self-loop term)
__global__ void agg_init_kernel(const float* __restrict__ h,
                                const float* __restrict__ dinv,
                                const float* __restrict__ bias,
                                float* __restrict__ out, int n) {
    int i = blockIdx.x * blockDim.x + threadIdx.x;
    if (i >= n * 20) return;
    int node = i / 20;
    int c    = i - node * 20;
    float di = dinv[node];
    out[i] = bias[c] + h[i] * di * di;
}

// out[dst] += h[src] * dinv[src]*dinv[dst]   (h rows are 80B -> 16B aligned)
__global__ void agg_edges_kernel(const float* __restrict__ h,
                                 const float* __restrict__ dinv,
                                 const int* __restrict__ src,
                                 const int* __restrict__ dst,
                                 float* __restrict__ out, int e) {
    int i = blockIdx.x * blockDim.x + threadIdx.x;
    if (i >= e) return;
    int s = src[i], d = dst[i];
    float w = dinv[s] * dinv[d];
    const float4* hs = (const float4*)(h + (size_t)s * 20);
    float* ob = out + (size_t)d * 20;
#pragma unroll
    for (int j = 0; j < 5; ++j) {
        float4 v = hs[j];
        atomicAdd(ob + j * 4 + 0, v.x * w);
        atomicAdd(ob + j * 4 + 1, v.y * w);
        atomicAdd(ob + j * 4 + 2, v.z * w);
        atomicAdd(ob + j * 4 + 3, v.w * w);
    }
}

__global__ void elu_kernel(float* __restrict__ a, int n) {
    int i = blockIdx.x * blockDim.x + threadIdx.x;
    if (i >= n) return;
    float v = a[i];
    a[i] = (v > 0.0f) ? v : expm1f(v);
}

// ---------------------------------------------------------------- decoder
// out = relu(h @ W1 + b1) @ W2 + b2      (20 -> 10 -> 1)
__global__ void decoder_kernel(const float* __restrict__ h,
                               const float* __restrict__ W1,
                               const float* __restrict__ b1,
                               const float* __restrict__ W2,
                               const float* __restrict__ b2,
                               float* __restrict__ out, int n) {
    int i = blockIdx.x * blockDim.x + threadIdx.x;
    if (i >= n) return;
    const float* hr = h + (size_t)i * 20;
    float z[10];
#pragma unroll
    for (int j = 0; j < 10; ++j) z[j] = b1[j];
#pragma unroll
    for (int c = 0; c < 20; ++c) {
        float hv = hr[c];
#pragma unroll
        for (int j = 0; j < 10; ++j) z[j] += hv * W1[c * 10 + j];
    }
    float acc = b2[0];
#pragma unroll
    for (int j = 0; j < 10; ++j) acc += fmaxf(z[j], 0.0f) * W2[j];
    out[i] = acc;
}

// ---------------------------------------------------------------- launch
extern "C" void kernel_launch(void* const* d_in, const int* in_sizes, int n_in,
                              void* d_out, int out_size, void* d_ws, size_t ws_size,
                              hipStream_t stream) {
    const float* x    = (const float*)d_in[0];
    const int*   ei   = (const int*)d_in[1];
    const float* W1   = (const float*)d_in[2];
    const float* b1   = (const float*)d_in[3];
    const float* W2   = (const float*)d_in[4];
    const float* b2   = (const float*)d_in[5];
    const float* dW1  = (const float*)d_in[6];
    const float* db1  = (const float*)d_in[7];
    const float* dW2  = (const float*)d_in[8];
    const float* db2  = (const float*)d_in[9];
    float* out = (float*)d_out;

    const int N = in_sizes[0] / 128;
    const int E = in_sizes[1] / 2;
    const int* src = ei;        // edge_index[0]
    const int* dst = ei + E;    // edge_index[1]

    float* ws   = (float*)d_ws;
    float* dinv = ws;                        // N floats (deg -> dinv in place)
    float* bufA = dinv + N;                  // N*20 floats
    float* bufB = bufA + (size_t)N * 20;     // N*20 floats

    const int NE = N * 20;
    dim3 blk(THREADS);
    dim3 gN((N + THREADS - 1) / THREADS);
    dim3 gE((E + THREADS - 1) / THREADS);
    dim3 gNE((NE + THREADS - 1) / THREADS);

    const int nTiles = (N + 15) / 16;
    dim3 gGemm((nTiles + WAVES_PER_BLOCK - 1) / WAVES_PER_BLOCK);

    // --- symmetric normalization: dinv = rsqrt(deg + self-loop)
    deg_init_kernel<<<gN, blk, 0, stream>>>(dinv, N);
    deg_accum_kernel<<<gE, blk, 0, stream>>>(dst, dinv, E);
    rsqrt_kernel<<<gN, blk, 0, stream>>>(dinv, N);

    // --- layer 1: h1 = x @ W1 ; agg ; +b1 ; ELU
    wmma_gemm_kernel<128><<<gGemm, blk, 0, stream>>>(x, W1, bufA, N, nTiles);
    agg_init_kernel<<<gNE, blk, 0, stream>>>(bufA, dinv, b1, bufB, N);
    agg_edges_kernel<<<gE, blk, 0, stream>>>(bufA, dinv, src, dst, bufB, E);
    elu_kernel<<<gNE, blk, 0, stream>>>(bufB, NE);

    // --- layer 2: h2 = h1 @ W2 ; agg ; +b2 ; ELU
    wmma_gemm_kernel<20><<<gGemm, blk, 0, stream>>>(bufB, W2, bufA, N, nTiles);
    agg_init_kernel<<<gNE, blk, 0, stream>>>(bufA, dinv, b2, bufB, N);
    agg_edges_kernel<<<gE, blk, 0, stream>>>(bufA, dinv, src, dst, bufB, E);
    elu_kernel<<<gNE, blk, 0, stream>>>(bufB, NE);

    // --- decoder
    decoder_kernel<<<gN, blk, 0, stream>>>(bufB, dW1, db1, dW2, db2, out, N);
}